// RGCN_40166534152455
// MI455X (gfx1250) — compile-verified
//
#include <hip/hip_runtime.h>
#include <cstdint>

// ---------------------------------------------------------------------------
// RGCN (2 layers, basis decomposition) for MI455X / gfx1250, wave32 + WMMA.
//
//   aggB[n, b*128+d] = sum_{e: dst=n} comp[type_e,b]*norm_e*H[src_e,d]
//   out = aggB @ reshape(basis, [512, d_out]) + bias   (fp32 WMMA 16x16x4)
//
// aggB (102.4 MB) is L2-resident (192 MB L2) -> scatter atomics resolve in L2.
// B matrix is pre-packed into WMMA fragment order and staged into LDS per
// workgroup with global_load_async_to_lds_b128 (ASYNCcnt path).
// ---------------------------------------------------------------------------

typedef __attribute__((ext_vector_type(2))) float v2f;
typedef __attribute__((ext_vector_type(8))) float v8f;

#define NNODES 50000
#define NEDGES 800000
#define NREL   8
#define NBASES 4
#define DFEAT  128                  // D_IN == D_HID == 128
#define DHID   128
#define DOUT   64
#define KDIM   (NBASES * DFEAT)     // 512, GEMM reduction dim for both layers
#define MTILES (NNODES / 16)        // 3125 exactly

// ---------------------------------------------------------------- zero fill
__global__ void rgcn_zero_f4(float4* __restrict__ p, long n4) {
  long i = (long)blockIdx.x * blockDim.x + threadIdx.x;
  if (i < n4) {
    float4 z; z.x = 0.f; z.y = 0.f; z.z = 0.f; z.w = 0.f;
    p[i] = z;
  }
}

// ------------------------------------------------------- edge scatter phase
// 32 threads per edge; each thread handles 4 contiguous channels (float4),
// scattering into 4 basis accumulators via fp32 atomics (L2-resident target).
__global__ __launch_bounds__(256)
void rgcn_scatter(const float* __restrict__ H,      // [N, 128]
                  const int*   __restrict__ esrc,   // [E]
                  const int*   __restrict__ edst,   // [E]
                  const int*   __restrict__ etype,  // [E]
                  const float* __restrict__ enorm,  // [E]
                  const float* __restrict__ comp,   // [R, B]
                  float*       __restrict__ agg)    // [N, 512]
{
  long idx = (long)blockIdx.x * blockDim.x + threadIdx.x;
  int e = (int)(idx >> 5);
  if (e >= NEDGES) return;
  int c4 = (int)(idx & 31) << 2;            // channel offset 0..124

  int   src = esrc[e];
  int   dst = edst[e];
  int   t   = etype[e];
  float nrm = enorm[e];

  const float4 h = *(const float4*)(H + (long)src * DFEAT + c4);

  float w[NBASES];
#pragma unroll
  for (int b = 0; b < NBASES; ++b) w[b] = comp[t * NBASES + b] * nrm;

  float* base = agg + (long)dst * KDIM + c4;
#pragma unroll
  for (int b = 0; b < NBASES; ++b) {
    float* p = base + b * DFEAT;
    atomicAdd(p + 0, w[b] * h.x);
    atomicAdd(p + 1, w[b] * h.y);
    atomicAdd(p + 2, w[b] * h.z);
    atomicAdd(p + 3, w[b] * h.w);
  }
}

// ------------------------------------------------ pack B into fragment order
// Bp[(k4*NT + t)*32 + lane] = { Bm[(4*k4+kh)*NO + col], Bm[(4*k4+kh+1)*NO + col] }
// with kh = (lane>>4)*2, col = t*16 + (lane&15).  One b64 load per fragment.
template <int NO>
__global__ void rgcn_packB(const float* __restrict__ Bm,  // [512, NO]
                           v2f*         __restrict__ Bp)
{
  constexpr int NT = NO / 16;
  int idx = blockIdx.x * blockDim.x + threadIdx.x;
  if (idx >= (KDIM / 4) * NT * 32) return;
  int lane = idx & 31;
  int t    = (idx >> 5) % NT;
  int k4   = idx / (32 * NT);
  int kh   = (lane >> 4) << 1;
  int col  = t * 16 + (lane & 15);
  v2f v;
  v.x = Bm[(4 * k4 + kh + 0) * NO + col];
  v.y = Bm[(4 * k4 + kh + 1) * NO + col];
  Bp[idx] = v;
}

// --------------------------------------------------------- WMMA f32 GEMM
// C[M, NO] = relu?( A[M,512] @ B[512,NO] + bias ), B pre-packed (Bp).
// 8 waves / block, one 16-row tile per wave; all waves share the B stream,
// which is staged chunk-by-chunk into LDS via async-to-LDS.
template <int NO, bool RELU>
__global__ __launch_bounds__(256)
void rgcn_gemm(const float* __restrict__ A,     // [M, 512]
               const v2f*   __restrict__ Bp,    // packed fragments
               const float* __restrict__ bias,  // [NO]
               float*       __restrict__ C,     // [M, NO]
               int M)
{
  constexpr int NT       = NO / 16;
  constexpr int CS       = 32;                 // k-steps per LDS chunk
  constexpr int CHUNKS   = (KDIM / 4) / CS;    // 4
  constexpr int CHUNK_V2 = CS * NT * 32;       // float2 elems per chunk
  constexpr int NITER    = (CHUNK_V2 * 8) / (256 * 16);  // b128 copies/thread

  __shared__ v2f sB[CHUNK_V2];                 // 64 KB (NO=128) / 32 KB (NO=64)

  const int lane = threadIdx.x & 31;
  const int wave = threadIdx.x >> 5;
  const int tile = blockIdx.x * 8 + wave;
  const int m0   = tile * 16;
  if (m0 >= M) return;   // wave-uniform; barrier completes on terminated waves

  const int r  = lane & 15;
  const int kh = (lane >> 4) << 1;

  v8f acc[NT];
#pragma unroll
  for (int t = 0; t < NT; ++t) acc[t] = (v8f){0.f,0.f,0.f,0.f,0.f,0.f,0.f,0.f};

  const float*   arow    = A + (long)(m0 + r) * KDIM + kh;
  const uint32_t ldsBase = (uint32_t)(uintptr_t)&sB[0];

  for (int c = 0; c < CHUNKS; ++c) {
    // ---- async stage: packed B chunk -> LDS (no VGPR round trip)
    const v2f* gsrc = Bp + c * CHUNK_V2;
#pragma unroll
    for (int i = 0; i < NITER; ++i) {
      uint32_t boff = (uint32_t)(threadIdx.x + i * 256) * 16u;
      asm volatile("global_load_async_to_lds_b128 %0, %1, %2 offset:0"
                   :: "v"(ldsBase + boff), "v"(boff), "s"(gsrc)
                   : "memory");
    }
    asm volatile("s_wait_asynccnt 0x0" ::: "memory");
    __syncthreads();

    // ---- compute: 32 k-steps, A from global (b64), B from LDS (ds b64)
#pragma unroll 4
    for (int s = 0; s < CS; ++s) {
      const int k = (c * CS + s) * 4;
      v2f a = *(const v2f*)(arow + k);         // 8B-aligned per-lane load
#pragma unroll
      for (int t = 0; t < NT; ++t) {
        v2f b = sB[(s * NT + t) * 32 + lane];  // ds_load_b64
        acc[t] = __builtin_amdgcn_wmma_f32_16x16x4_f32(
            false, a, false, b, (short)0, acc[t], false, false);
      }
    }
    __syncthreads();
  }

  // C/D layout: VGPR v, lanes 0-15 -> M = v; lanes 16-31 -> M = v + 8; N = r
  const int rowhi = (lane >> 4) * 8;
#pragma unroll
  for (int t = 0; t < NT; ++t) {
    float bb = bias[t * 16 + r];
#pragma unroll
    for (int v = 0; v < 8; ++v) {
      float val = acc[t][v] + bb;
      if (RELU) val = fmaxf(val, 0.f);
      C[(long)(m0 + rowhi + v) * NO + t * 16 + r] = val;
    }
  }
}

// ---------------------------------------------------------------- launcher
extern "C" void kernel_launch(void* const* d_in, const int* in_sizes, int n_in,
                              void* d_out, int out_size, void* d_ws, size_t ws_size,
                              hipStream_t stream)
{
  const float* X      = (const float*)d_in[0];
  const int*   esrc   = (const int*)  d_in[1];
  const int*   edst   = (const int*)  d_in[2];
  const int*   etype  = (const int*)  d_in[3];
  const float* enorm  = (const float*)d_in[4];
  const float* basis0 = (const float*)d_in[5];   // [4,128,128] == [512,128]
  const float* comp0  = (const float*)d_in[6];   // [8,4]
  const float* bias0  = (const float*)d_in[7];   // [128]
  const float* basis1 = (const float*)d_in[8];   // [4,128,64] == [512,64]
  const float* comp1  = (const float*)d_in[9];   // [8,4]
  const float* bias1  = (const float*)d_in[10];  // [64]

  float* agg = (float*)d_ws;                     // [N, 512] = 102.4 MB
  float* H1  = agg + (size_t)NNODES * KDIM;      // [N, 128] =  25.6 MB
  v2f*   Bp0 = (v2f*)(H1 + (size_t)NNODES * DFEAT);            // 256 KB
  v2f*   Bp1 = Bp0 + (size_t)(KDIM / 4) * (DHID / 16) * 32;    // 128 KB

  const long n4 = (long)NNODES * KDIM / 4;
  const int  zgrid = (int)((n4 + 255) / 256);
  const long sthreads = (long)NEDGES * 32;
  const int  sgrid = (int)((sthreads + 255) / 256);
  const int  ggrid = (MTILES + 7) / 8;

  // ---- one-time (per call) B repack into WMMA fragment order
  rgcn_packB<DHID><<<((KDIM/4)*(DHID/16)*32 + 255) / 256, 256, 0, stream>>>(basis0, Bp0);
  rgcn_packB<DOUT><<<((KDIM/4)*(DOUT/16)*32 + 255) / 256, 256, 0, stream>>>(basis1, Bp1);

  // ---- layer 0: X --scatter--> agg --gemm+relu--> H1
  rgcn_zero_f4<<<zgrid, 256, 0, stream>>>((float4*)agg, n4);
  rgcn_scatter<<<sgrid, 256, 0, stream>>>(X, esrc, edst, etype, enorm, comp0, agg);
  rgcn_gemm<DHID, true><<<ggrid, 256, 0, stream>>>(agg, Bp0, bias0, H1, NNODES);

  // ---- layer 1: H1 --scatter--> agg --gemm--> out
  rgcn_zero_f4<<<zgrid, 256, 0, stream>>>((float4*)agg, n4);
  rgcn_scatter<<<sgrid, 256, 0, stream>>>(H1, esrc, edst, etype, enorm, comp1, agg);
  rgcn_gemm<DOUT, false><<<ggrid, 256, 0, stream>>>(agg, Bp1, bias1, (float*)d_out, NNODES);
}